// PointNetBasis_feat_40699110097103
// MI455X (gfx1250) — compile-verified
//
#include <hip/hip_runtime.h>
#include <hip/hip_bf16.h>
#include <math.h>

typedef __attribute__((ext_vector_type(2))) float v2f;
typedef __attribute__((ext_vector_type(8))) float v8f;

#define BN_EPS 1e-5f

// ---------------------------------------------------------------------------
// Input transpose: xyz (B,3,N) -> x0 (B,N,3)
// ---------------------------------------------------------------------------
__global__ void k_transpose_in(const float* __restrict__ xyz, int B, int N,
                               float* __restrict__ x0) {
  long long i = (long long)blockIdx.x * blockDim.x + threadIdx.x;
  long long total = (long long)B * N * 3;
  if (i >= total) return;
  int d = (int)(i % 3);
  long long t = i / 3;
  int n = (int)(t % N);
  int b = (int)(t / N);
  x0[i] = xyz[((long long)b * 3 + d) * N + n];
}

// ---------------------------------------------------------------------------
// Farthest point sampling: one block per batch, dist kept in LDS.
// ---------------------------------------------------------------------------
#define FPS_MAXN 8192
__global__ void k_fps(const float* __restrict__ xyz, int N, int npoint,
                      int* __restrict__ fidx) {
  __shared__ float dist[FPS_MAXN];
  __shared__ float rv[256];
  __shared__ int   ri[256];
  __shared__ float cx, cy, cz;
  __shared__ int   cur;
  int b = blockIdx.x;
  int t = threadIdx.x;
  const float* p = xyz + (long long)b * N * 3;
  for (int i = t; i < N; i += 256) dist[i] = 1e10f;
  if (t == 0) cur = 0;
  __syncthreads();
  for (int s = 0; s < npoint; ++s) {
    if (t == 0) {
      int f = cur;
      fidx[b * npoint + s] = f;
      cx = p[f * 3 + 0]; cy = p[f * 3 + 1]; cz = p[f * 3 + 2];
    }
    __syncthreads();
    float bm = -1.0f; int bi = 0x7fffffff;
    for (int i = t; i < N; i += 256) {
      float dx = p[i * 3 + 0] - cx;
      float dy = p[i * 3 + 1] - cy;
      float dz = p[i * 3 + 2] - cz;
      float d = dx * dx + dy * dy + dz * dz;
      d = fminf(dist[i], d);
      dist[i] = d;
      if (d > bm) { bm = d; bi = i; }
    }
    rv[t] = bm; ri[t] = bi;
    __syncthreads();
    for (int off = 128; off > 0; off >>= 1) {
      if (t < off) {
        if (rv[t + off] > rv[t] ||
            (rv[t + off] == rv[t] && ri[t + off] < ri[t])) {
          rv[t] = rv[t + off]; ri[t] = ri[t + off];
        }
      }
      __syncthreads();
    }
    if (t == 0) cur = ri[0];
    __syncthreads();
  }
}

// ---------------------------------------------------------------------------
// Gather sampled centroids
// ---------------------------------------------------------------------------
__global__ void k_gather_xyz(const float* __restrict__ xyz, const int* __restrict__ fidx,
                             int B, int N, int S, float* __restrict__ nx) {
  int i = blockIdx.x * blockDim.x + threadIdx.x;
  if (i >= B * S) return;
  int b = i / S;
  int f = fidx[i];
  const float* src = xyz + ((long long)b * N + f) * 3;
  nx[i * 3 + 0] = src[0];
  nx[i * 3 + 1] = src[1];
  nx[i * 3 + 2] = src[2];
}

// ---------------------------------------------------------------------------
// Ball query: first `ns` indices with d^2 <= r^2 in ascending order, padded
// with the first hit (matches sort+truncate+where in the reference).
// ---------------------------------------------------------------------------
__global__ void k_ball_query(const float* __restrict__ xyz, const float* __restrict__ nx,
                             int B, int N, int S, int ns, float r2,
                             int* __restrict__ gidx) {
  int i = blockIdx.x * blockDim.x + threadIdx.x;
  if (i >= B * S) return;
  int b = i / S;
  const float* p = xyz + (long long)b * N * 3;
  float qx = nx[i * 3 + 0], qy = nx[i * 3 + 1], qz = nx[i * 3 + 2];
  int* out = gidx + (long long)i * ns;
  int cnt = 0, first = 0;
  for (int j = 0; j < N && cnt < ns; ++j) {
    float dx = p[j * 3 + 0] - qx;
    float dy = p[j * 3 + 1] - qy;
    float dz = p[j * 3 + 2] - qz;
    float d = dx * dx + dy * dy + dz * dz;
    if (d <= r2) {
      if (cnt == 0) first = j;
      out[cnt++] = j;
    }
  }
  for (int k = cnt; k < ns; ++k) out[k] = first;
}

// ---------------------------------------------------------------------------
// Build grouped input rows: concat(grouped_xyz - center, grouped_points)
// ---------------------------------------------------------------------------
__global__ void k_group_concat(const float* __restrict__ xyz, const float* __restrict__ pts,
                               const float* __restrict__ nx, const int* __restrict__ gidx,
                               int B, int N, int S, int ns, int C,
                               float* __restrict__ xin) {
  int cin = 3 + C;
  long long total = (long long)B * S * ns * cin;
  long long i = (long long)blockIdx.x * blockDim.x + threadIdx.x;
  if (i >= total) return;
  long long r = i / cin;
  int c = (int)(i % cin);
  int k = (int)(r % ns);
  long long g = r / ns;           // g = b*S + s
  int b = (int)(g / S);
  int j = gidx[g * ns + k];
  if (c < 3) {
    xin[i] = xyz[((long long)b * N + j) * 3 + c] - nx[g * 3 + c];
  } else {
    xin[i] = pts[((long long)b * N + j) * C + (c - 3)];
  }
}

// ---------------------------------------------------------------------------
// GEMM + bias via V_WMMA_F32_16X16X4_F32.  Y[r,o] = sum_c X[r,c]*W[o,c] + b[o]
// One wave per 16x16 output tile.  Main K loop is branch-free: out-of-range
// rows are clamped (duplicate reads, results discarded at store) and the
// K tail (cin % 4 != 0) is peeled into a single guarded iteration.
// ---------------------------------------------------------------------------
__global__ void k_gemm_bias_wmma(const float* __restrict__ X, const float* __restrict__ W,
                                 const float* __restrict__ bias, float* __restrict__ Y,
                                 int nrows, int cin, int cout) {
  int rtiles = (nrows + 15) >> 4;
  int ctiles = cout >> 4;
  int wave = (blockIdx.x * blockDim.x + threadIdx.x) >> 5;
  int lane = threadIdx.x & 31;
  if (wave >= rtiles * ctiles) return;   // uniform per wave
  int rt = wave / ctiles;
  int ct = wave % ctiles;
  int m = lane & 15;
  int half = lane >> 4;                   // ISA layout: lanes 16-31 hold K+2,K+3
  int row = rt * 16 + m;
  int col = ct * 16 + m;
  int rowL = (row < nrows) ? row : (nrows - 1);  // clamp; extra rows never stored
  const float* xr = X + (long long)rowL * cin + half * 2;
  const float* wr = W + (long long)col * cin + half * 2;
  v8f acc = {};
  int kmain = cin & ~3;
  int k = 0;
  #pragma unroll 2
  for (; k < kmain; k += 4) {           // unconditional loads: k + half*2 + 1 < cin
    v2f a, b;
    a.x = xr[k];
    a.y = xr[k + 1];
    b.x = wr[k];
    b.y = wr[k + 1];
    acc = __builtin_amdgcn_wmma_f32_16x16x4_f32(false, a, false, b,
                                                (short)0, acc, false, false);
  }
  if (k < cin) {                         // single guarded tail iteration
    int k0 = k + half * 2;
    v2f a = {0.0f, 0.0f};
    v2f b = {0.0f, 0.0f};
    if (k0 < cin)     { a.x = xr[k];     b.x = wr[k]; }
    if (k0 + 1 < cin) { a.y = xr[k + 1]; b.y = wr[k + 1]; }
    acc = __builtin_amdgcn_wmma_f32_16x16x4_f32(false, a, false, b,
                                                (short)0, acc, false, false);
  }
  int n = ct * 16 + m;
  float bv = bias[n];
  #pragma unroll
  for (int v = 0; v < 8; ++v) {
    int r = rt * 16 + v + 8 * half;
    if (r < nrows) Y[(long long)r * cout + n] = acc[v] + bv;
  }
}

// ---------------------------------------------------------------------------
// Batch-norm statistics (population mean/var per channel) + normalize/ReLU
// ---------------------------------------------------------------------------
__global__ void k_bn_stats(const float* __restrict__ X, int nrows, int cout,
                           float* __restrict__ mean, float* __restrict__ var) {
  __shared__ float s1[256];
  __shared__ float s2[256];
  int c = blockIdx.x;
  int t = threadIdx.x;
  float a = 0.0f, b = 0.0f;
  for (int r = t; r < nrows; r += 256) {
    float v = X[(long long)r * cout + c];
    a += v; b += v * v;
  }
  s1[t] = a; s2[t] = b;
  __syncthreads();
  for (int off = 128; off > 0; off >>= 1) {
    if (t < off) { s1[t] += s1[t + off]; s2[t] += s2[t + off]; }
    __syncthreads();
  }
  if (t == 0) {
    float m = s1[0] / (float)nrows;
    mean[c] = m;
    var[c] = fmaxf(s2[0] / (float)nrows - m * m, 0.0f);
  }
}

__global__ void k_bn_relu(float* __restrict__ X, const float* __restrict__ mean,
                          const float* __restrict__ var, const float* __restrict__ gamma,
                          const float* __restrict__ beta, long long total, int cout) {
  long long i = (long long)blockIdx.x * blockDim.x + threadIdx.x;
  if (i >= total) return;
  int c = (int)(i % cout);
  float y = (X[i] - mean[c]) * rsqrtf(var[c] + BN_EPS) * gamma[c] + beta[c];
  X[i] = fmaxf(y, 0.0f);
}

// ---------------------------------------------------------------------------
// Max over the sample axis: X (groups, ns, C) -> Y (groups, C)
// ---------------------------------------------------------------------------
__global__ void k_maxpool_ns(const float* __restrict__ X, int groups, int ns, int C,
                             float* __restrict__ Y) {
  long long i = (long long)blockIdx.x * blockDim.x + threadIdx.x;
  if (i >= (long long)groups * C) return;
  long long g = i / C;
  int c = (int)(i % C);
  float m = -1e30f;
  for (int k = 0; k < ns; ++k) m = fmaxf(m, X[(g * ns + k) * C + c]);
  Y[i] = m;
}

// ---------------------------------------------------------------------------
// 3-NN with inverse-distance weights (matches top_k(-d,3) tie behavior)
// ---------------------------------------------------------------------------
__global__ void k_three_nn(const float* __restrict__ x1, const float* __restrict__ x2,
                           int B, int N1, int N2,
                           int* __restrict__ idx3, float* __restrict__ w3) {
  int i = blockIdx.x * blockDim.x + threadIdx.x;
  if (i >= B * N1) return;
  int b = i / N1;
  const float* q = x1 + (long long)i * 3;
  const float* p = x2 + (long long)b * N2 * 3;
  float qx = q[0], qy = q[1], qz = q[2];
  float d0 = 1e30f, d1 = 1e30f, d2 = 1e30f;
  int i0 = 0, i1 = 0, i2 = 0;
  for (int j = 0; j < N2; ++j) {
    float dx = p[j * 3 + 0] - qx;
    float dy = p[j * 3 + 1] - qy;
    float dz = p[j * 3 + 2] - qz;
    float d = dx * dx + dy * dy + dz * dz;
    if (d < d0)      { d2 = d1; i2 = i1; d1 = d0; i1 = i0; d0 = d; i0 = j; }
    else if (d < d1) { d2 = d1; i2 = i1; d1 = d;  i1 = j; }
    else if (d < d2) { d2 = d;  i2 = j; }
  }
  float r0 = 1.0f / (d0 + 1e-8f);
  float r1 = 1.0f / (d1 + 1e-8f);
  float r2 = 1.0f / (d2 + 1e-8f);
  float s = r0 + r1 + r2;
  idx3[i * 3 + 0] = i0; idx3[i * 3 + 1] = i1; idx3[i * 3 + 2] = i2;
  w3[i * 3 + 0] = r0 / s; w3[i * 3 + 1] = r1 / s; w3[i * 3 + 2] = r2 / s;
}

// ---------------------------------------------------------------------------
// Build FP input rows: concat(points1, interpolated points2)
// ---------------------------------------------------------------------------
__global__ void k_fp_build_xin(const float* __restrict__ p1, int C1,
                               const float* __restrict__ p2, int C2,
                               const int* __restrict__ idx3, const float* __restrict__ w3,
                               int B, int N1, int N2, float* __restrict__ xin) {
  int cin = C1 + C2;
  long long total = (long long)B * N1 * cin;
  long long i = (long long)blockIdx.x * blockDim.x + threadIdx.x;
  if (i >= total) return;
  long long r = i / cin;
  int c = (int)(i % cin);
  if (c < C1) {
    xin[i] = p1[r * C1 + c];
  } else {
    int cc = c - C1;
    int b = (int)(r / N1);
    const int* id = idx3 + r * 3;
    const float* w = w3 + r * 3;
    float v = 0.0f;
    #pragma unroll
    for (int k = 0; k < 3; ++k)
      v += w[k] * p2[((long long)b * N2 + id[k]) * C2 + cc];
    xin[i] = v;
  }
}

// ---------------------------------------------------------------------------
// Final: out[b,c,n] = (c<128) ? head[b,c] : pointfeat[b,n,c-128]
// ---------------------------------------------------------------------------
__global__ void k_final(const float* __restrict__ h2, const float* __restrict__ pf,
                        int B, int N, float* __restrict__ out) {
  long long total = (long long)B * 256 * N;
  long long i = (long long)blockIdx.x * blockDim.x + threadIdx.x;
  if (i >= total) return;
  int n = (int)(i % N);
  long long t = i / N;
  int c = (int)(t % 256);
  int b = (int)(t / 256);
  out[i] = (c < 128) ? h2[b * 128 + c]
                     : pf[((long long)b * N + n) * 128 + (c - 128)];
}

// ---------------------------------------------------------------------------
// Host orchestration
// ---------------------------------------------------------------------------
struct Lyr { const float *W, *b, *g, *be; int cin, cout; };

static inline int nblk(long long n) { return (int)((n + 255) / 256); }

static float* run_mlp(const float* xin, int nrows, const Lyr* ls, int nl,
                      float* bufA, float* bufB, float* mean, float* var,
                      hipStream_t st) {
  const float* cur = xin;
  float* out = nullptr;
  for (int i = 0; i < nl; ++i) {
    out = (i % 2 == 0) ? bufA : bufB;
    int cin = ls[i].cin, cout = ls[i].cout;
    long long waves = (long long)((nrows + 15) / 16) * (cout / 16);
    k_gemm_bias_wmma<<<(int)((waves + 7) / 8), 256, 0, st>>>(
        cur, ls[i].W, ls[i].b, out, nrows, cin, cout);
    k_bn_stats<<<cout, 256, 0, st>>>(out, nrows, cout, mean, var);
    long long tot = (long long)nrows * cout;
    k_bn_relu<<<nblk(tot), 256, 0, st>>>(out, mean, var, ls[i].g, ls[i].be, tot, cout);
    cur = out;
  }
  return out;
}

static void sa_stage(const float* xyz, const float* pts, int B, int N, int C,
                     int S, float radius, int ns, const Lyr* ls, int nl,
                     float* nx, float* np, int* fidx, int* gidx,
                     float* bufA, float* bufB, float* mean, float* var,
                     hipStream_t st) {
  k_fps<<<B, 256, 0, st>>>(xyz, N, S, fidx);
  k_gather_xyz<<<nblk((long long)B * S), 256, 0, st>>>(xyz, fidx, B, N, S, nx);
  k_ball_query<<<nblk((long long)B * S), 256, 0, st>>>(xyz, nx, B, N, S, ns,
                                                       radius * radius, gidx);
  int cin = 3 + C;
  long long rows = (long long)B * S * ns;
  k_group_concat<<<nblk(rows * cin), 256, 0, st>>>(xyz, pts, nx, gidx,
                                                   B, N, S, ns, C, bufB);
  float* act = run_mlp(bufB, (int)rows, ls, nl, bufA, bufB, mean, var, st);
  int Co = ls[nl - 1].cout;
  k_maxpool_ns<<<nblk((long long)B * S * Co), 256, 0, st>>>(act, B * S, ns, Co, np);
}

static float* fp_stage(const float* x1, const float* x2, const float* p1, int C1,
                       const float* p2, int C2, int B, int N1, int N2,
                       const Lyr* ls, int nl, int* idx3, float* w3,
                       float* bufA, float* bufB, float* mean, float* var,
                       hipStream_t st) {
  k_three_nn<<<nblk((long long)B * N1), 256, 0, st>>>(x1, x2, B, N1, N2, idx3, w3);
  int cin = C1 + C2;
  long long tot = (long long)B * N1 * cin;
  k_fp_build_xin<<<nblk(tot), 256, 0, st>>>(p1, C1, p2, C2, idx3, w3, B, N1, N2, bufB);
  return run_mlp(bufB, B * N1, ls, nl, bufA, bufB, mean, var, st);
}

extern "C" void kernel_launch(void* const* d_in, const int* in_sizes, int n_in,
                              void* d_out, int out_size, void* d_ws, size_t ws_size,
                              hipStream_t stream) {
  const int B = 8, N = 8192;

  // ---- map inputs (handles insertion order and jax-sorted tree order) ----
  const float* P[88];
  const float* xyz_in;
  if (in_sizes[0] == B * 3 * N) {
    xyz_in = (const float*)d_in[0];
    for (int i = 0; i < 88; ++i) P[i] = (const float*)d_in[1 + i];
  } else {
    xyz_in = (const float*)d_in[n_in - 1];
    // sorted keys: dense1,dense2,fp1,fp2,fp3,fp4,sa1,sa2,sa3,sa4
    const int base[10]  = {80, 84, 72, 64, 56, 48, 0, 12, 24, 36};
    const int count[10] = {4, 4, 8, 8, 8, 8, 12, 12, 12, 12};
    int src = 0;
    for (int g = 0; g < 10; ++g)
      for (int j = 0; j < count[g]; ++j)
        P[base[g] + j] = (const float*)d_in[src++];
  }

  auto mk = [&](int canon, int li, int cin, int cout) -> Lyr {
    Lyr L;
    L.W = P[canon + 4 * li + 0]; L.b = P[canon + 4 * li + 1];
    L.g = P[canon + 4 * li + 2]; L.be = P[canon + 4 * li + 3];
    L.cin = cin; L.cout = cout;
    return L;
  };
  Lyr sa1[3] = { mk(0, 0, 6, 32),    mk(0, 1, 32, 32),   mk(0, 2, 32, 64) };
  Lyr sa2[3] = { mk(12, 0, 67, 64),  mk(12, 1, 64, 64),  mk(12, 2, 64, 128) };
  Lyr sa3[3] = { mk(24, 0, 131, 128),mk(24, 1, 128, 128),mk(24, 2, 128, 256) };
  Lyr sa4[3] = { mk(36, 0, 259, 256),mk(36, 1, 256, 256),mk(36, 2, 256, 512) };
  Lyr fp4[2] = { mk(48, 0, 768, 256), mk(48, 1, 256, 256) };
  Lyr fp3[2] = { mk(56, 0, 384, 256), mk(56, 1, 256, 256) };
  Lyr fp2[2] = { mk(64, 0, 320, 256), mk(64, 1, 256, 128) };
  Lyr fp1[2] = { mk(72, 0, 128, 128), mk(72, 1, 128, 128) };
  Lyr dn1 = mk(80, 0, 512, 256);
  Lyr dn2 = mk(84, 0, 256, 128);

  // ---- workspace layout ----
  char* w = (char*)d_ws;
  auto alloc = [&](size_t nbytes) -> void* {
    void* p = (void*)w;
    w += (nbytes + 255) & ~(size_t)255;
    return p;
  };
  float* x0   = (float*)alloc((size_t)B * N * 3 * 4);        // 196608
  float* l1x  = (float*)alloc((size_t)B * 512 * 3 * 4);
  float* l1p  = (float*)alloc((size_t)B * 512 * 64 * 4);
  float* l2x  = (float*)alloc((size_t)B * 256 * 3 * 4);
  float* l2p  = (float*)alloc((size_t)B * 256 * 128 * 4);
  float* l3x  = (float*)alloc((size_t)B * 64 * 3 * 4);
  float* l3p  = (float*)alloc((size_t)B * 64 * 256 * 4);
  float* l4x  = (float*)alloc((size_t)B * 16 * 3 * 4);
  float* l4p  = (float*)alloc((size_t)B * 16 * 512 * 4);
  float* l3pf = (float*)alloc((size_t)B * 64 * 256 * 4);     // fp4 out
  float* l2pf = (float*)alloc((size_t)B * 256 * 256 * 4);    // fp3 out
  float* l1pf = (float*)alloc((size_t)B * 512 * 128 * 4);    // fp2 out
  float* bufA = (float*)alloc((size_t)8388608 * 4);
  float* bufB = (float*)alloc((size_t)8388608 * 4);
  float* mean = (float*)alloc(512 * 4);
  float* var  = (float*)alloc(512 * 4);
  float* gbuf = (float*)alloc((size_t)B * 512 * 4);
  float* dA   = (float*)alloc((size_t)B * 256 * 4);
  float* dB   = (float*)alloc((size_t)B * 256 * 4);
  int*   fidx = (int*)alloc((size_t)B * 512 * 4);
  int*   gidx = (int*)alloc((size_t)B * 512 * 32 * 4);
  int*   idx3 = (int*)alloc((size_t)B * N * 3 * 4);
  float* w3   = (float*)alloc((size_t)B * N * 3 * 4);
  (void)ws_size;

  // ---- pipeline ----
  k_transpose_in<<<nblk((long long)B * N * 3), 256, 0, stream>>>(xyz_in, B, N, x0);

  sa_stage(x0,  x0,  B, N,   3,   512, 0.025f, 32, sa1, 3, l1x, l1p, fidx, gidx, bufA, bufB, mean, var, stream);
  sa_stage(l1x, l1p, B, 512, 64,  256, 0.05f,  32, sa2, 3, l2x, l2p, fidx, gidx, bufA, bufB, mean, var, stream);
  sa_stage(l2x, l2p, B, 256, 128, 64,  0.1f,   32, sa3, 3, l3x, l3p, fidx, gidx, bufA, bufB, mean, var, stream);
  sa_stage(l3x, l3p, B, 64,  256, 16,  0.2f,   32, sa4, 3, l4x, l4p, fidx, gidx, bufA, bufB, mean, var, stream);

  // dense head: g = max over axis 1 of l4p (B,16,512), then two BN+ReLU denses
  k_maxpool_ns<<<nblk((long long)B * 512), 256, 0, stream>>>(l4p, B, 16, 512, gbuf);
  float* h1 = run_mlp(gbuf, B, &dn1, 1, dA, dB, mean, var, stream);   // -> dA (B,256)
  float* h2 = run_mlp(h1,   B, &dn2, 1, dB, dA, mean, var, stream);   // -> dB (B,128)

  // feature propagation
  float* a4 = fp_stage(l3x, l4x, l3p, 256, l4p, 512, B, 64, 16,
                       fp4, 2, idx3, w3, bufA, bufB, mean, var, stream);
  hipMemcpyAsync(l3pf, a4, (size_t)B * 64 * 256 * 4, hipMemcpyDeviceToDevice, stream);

  float* a3 = fp_stage(l2x, l3x, l2p, 128, l3pf, 256, B, 256, 64,
                       fp3, 2, idx3, w3, bufA, bufB, mean, var, stream);
  hipMemcpyAsync(l2pf, a3, (size_t)B * 256 * 256 * 4, hipMemcpyDeviceToDevice, stream);

  float* a2 = fp_stage(l1x, l2x, l1p, 64, l2pf, 256, B, 512, 256,
                       fp2, 2, idx3, w3, bufA, bufB, mean, var, stream);
  hipMemcpyAsync(l1pf, a2, (size_t)B * 512 * 128 * 4, hipMemcpyDeviceToDevice, stream);

  float* pf = fp_stage(x0, l1x, nullptr, 0, l1pf, 128, B, N, 512,
                       fp1, 2, idx3, w3, bufA, bufB, mean, var, stream);

  // final concat + transpose -> (B, 256, N)
  k_final<<<nblk((long long)B * 256 * N), 256, 0, stream>>>(h2, pf, B, N, (float*)d_out);
  (void)out_size;
}